// SupConLoss_43267500540145
// MI455X (gfx1250) — compile-verified
//
#include <hip/hip_runtime.h>
#include <hip/hip_bf16.h>
#include <math.h>

typedef __attribute__((ext_vector_type(2))) float v2f;
typedef __attribute__((ext_vector_type(8))) float v8f;

#define M_DIM 8192
#define K_DIM 128
#define INV_T 5.0f        // 1 / 0.2
#define NTILES (M_DIM / 16)   // 512 column tiles
#define KSTEPS (K_DIM / 4)    // 32 wmma_f32_16x16x4 steps

__global__ void supcon_zero_kernel(float* acc) {
    if (threadIdx.x == 0 && blockIdx.x == 0) acc[0] = 0.0f;
}

__global__ __launch_bounds__(256) void supcon_main_kernel(
        const float* __restrict__ F,
        const int*   __restrict__ labels,
        float*       __restrict__ accum) {
    const int lane  = threadIdx.x & 31;
    const int wave  = threadIdx.x >> 5;
    const int strip = blockIdx.x * 8 + wave;     // 0..511 (16-row strip)
    const int row_base = strip * 16;
    const int half = lane >> 4;                  // 0: M=0..7, 1: M=8..15
    const int l16  = lane & 15;

    // ---- A fragments: rows (row_base + l16), K striped per ISA 16x4 f32 layout
    // lane 0-15: K = 4*ks + {0,1}; lane 16-31: K = 4*ks + {2,3}
    v2f A[KSTEPS];
    const float* arow = F + (size_t)(row_base + l16) * K_DIM;
    #pragma unroll
    for (int ks = 0; ks < KSTEPS; ++ks)
        A[ks] = *(const v2f*)(arow + ks * 4 + half * 2);

    // row labels for the 8 C slots this lane owns (M = v + 8*half)
    int lab_row[8];
    #pragma unroll
    for (int v = 0; v < 8; ++v)
        lab_row[v] = labels[row_base + half * 8 + v];

    // online per-row partials (this lane sees columns  col ≡ l16 (mod 16))
    float mrow[8], Zrow[8], Srow[8], Prow[8];
    #pragma unroll
    for (int v = 0; v < 8; ++v) {
        mrow[v] = -INFINITY; Zrow[v] = 0.0f; Srow[v] = 0.0f; Prow[v] = 0.0f;
    }

    for (int tj = 0; tj < NTILES; ++tj) {
        const int col = tj * 16 + l16;           // this lane's column (B/C layout)
        const float* brow = F + (size_t)col * K_DIM;

        // C = A * B (+C), B fragment loads are symmetric to A since B = F^T
        v8f C = {};
        #pragma unroll
        for (int ks = 0; ks < KSTEPS; ++ks) {
            v2f b = *(const v2f*)(brow + ks * 4 + half * 2);
            C = __builtin_amdgcn_wmma_f32_16x16x4_f32(
                    /*neg_a=*/false, A[ks], /*neg_b=*/false, b,
                    /*c_mod=*/(short)0, C, /*reuse_a=*/false, /*reuse_b=*/false);
        }

        const int lab_col = labels[col];

        #pragma unroll
        for (int v = 0; v < 8; ++v) {
            const float x = C[v] * INV_T;                 // sim[row, col]
            const int row_g = row_base + v + 8 * half;
            const bool isdiag = (col == row_g);
            // row max includes the diagonal; Z excludes it (logits_mask)
            const float mnew = fmaxf(mrow[v], x);
            const float zadd = isdiag ? 0.0f : __expf(x - mnew);
            Zrow[v] = Zrow[v] * __expf(mrow[v] - mnew) + zadd;
            mrow[v] = mnew;
            const bool match = (!isdiag) && (lab_col == lab_row[v]);
            Srow[v] += match ? x : 0.0f;                  // raw sum over matched pairs
            Prow[v] += match ? 1.0f : 0.0f;
        }
    }

    // ---- merge online partials across the 16 lanes of each half (wave32)
    #pragma unroll
    for (int off = 8; off >= 1; off >>= 1) {
        #pragma unroll
        for (int v = 0; v < 8; ++v) {
            const float m2 = __shfl_xor(mrow[v], off, 32);
            const float Z2 = __shfl_xor(Zrow[v], off, 32);
            const float S2 = __shfl_xor(Srow[v], off, 32);
            const float P2 = __shfl_xor(Prow[v], off, 32);
            const float mm = fmaxf(mrow[v], m2);
            Zrow[v] = Zrow[v] * __expf(mrow[v] - mm) + Z2 * __expf(m2 - mm);
            mrow[v] = mm;
            Srow[v] += S2;
            Prow[v] += P2;
        }
    }

    // lanes 0 and 16 now hold final stats for rows {base..base+7} / {base+8..base+15}
    if (l16 == 0) {
        float acc = 0.0f;
        #pragma unroll
        for (int v = 0; v < 8; ++v) {
            const float logZ = __logf(Zrow[v]);
            // per_label_log_prob = (S - P*(m + logZ)) / (P + 1);  loss_i = -per
            const float per = (Srow[v] - Prow[v] * (mrow[v] + logZ)) / (Prow[v] + 1.0f);
            acc -= per;
        }
        atomicAdd(accum, acc);
    }
}

__global__ void supcon_finalize_kernel(const float* acc, float* out) {
    if (threadIdx.x == 0 && blockIdx.x == 0)
        out[0] = acc[0] * (1.0f / (float)M_DIM);
}

extern "C" void kernel_launch(void* const* d_in, const int* in_sizes, int n_in,
                              void* d_out, int out_size, void* d_ws, size_t ws_size,
                              hipStream_t stream) {
    const float* features = (const float*)d_in[0];
    const int*   labels   = (const int*)d_in[1];
    float* out = (float*)d_out;
    float* acc = (float*)d_ws;

    supcon_zero_kernel<<<1, 32, 0, stream>>>(acc);
    // 512 strips of 16 rows; 8 waves (256 threads) per block -> 64 blocks
    supcon_main_kernel<<<M_DIM / 16 / 8, 256, 0, stream>>>(features, labels, acc);
    supcon_finalize_kernel<<<1, 32, 0, stream>>>(acc, out);
}